// WassersteinLoss_62328565399961
// MI455X (gfx1250) — compile-verified
//
#include <hip/hip_runtime.h>
#include <math.h>

#define N 4096
#define D 512
#define LOGN 8.31776616671934f   // log(4096)

typedef float v2f __attribute__((ext_vector_type(2)));
typedef float v8f __attribute__((ext_vector_type(8)));

// ---------------- row squared norms: out[i] = sum_k x[i,k]^2 ----------------
__global__ void sqnorm_kernel(const float* __restrict__ x, float* __restrict__ out) {
    __shared__ float red[256];
    int row = blockIdx.x;
    int tid = threadIdx.x;
    const float* r = x + (size_t)row * D;
    float a0 = r[tid], a1 = r[tid + 256];
    red[tid] = a0 * a0 + a1 * a1;
    __syncthreads();
    for (int off = 128; off > 0; off >>= 1) {
        if (tid < off) red[tid] += red[tid + off];
        __syncthreads();
    }
    if (tid == 0) out[row] = red[0];
}

// ---------------- cost matrix via fp32 WMMA: M = max(x2+y2-2*X@Y^T, 0) ------
// Block = 256 threads = 8 waves; wave tile 32x32 (2x2 of 16x16); block tile 128x64.
__global__ __launch_bounds__(256) void cost_gemm_kernel(
    const float* __restrict__ X, const float* __restrict__ Y,
    const float* __restrict__ x2, const float* __restrict__ y2,
    float* __restrict__ M)
{
    int tid  = threadIdx.x;
    int lane = tid & 31;
    int wave = tid >> 5;           // 0..7
    int wm = wave & 3;             // 4 waves along M
    int wn = wave >> 2;            // 2 waves along N
    int m0 = blockIdx.x * 128 + wm * 32;
    int n0 = blockIdx.y * 64  + wn * 32;

    int l15 = lane & 15;
    int h2  = (lane >> 4) * 2;     // K offset: 0 or 2 per lane half

    const float* arow0 = X + (size_t)(m0      + l15) * D;
    const float* arow1 = X + (size_t)(m0 + 16 + l15) * D;
    const float* brow0 = Y + (size_t)(n0      + l15) * D;
    const float* brow1 = Y + (size_t)(n0 + 16 + l15) * D;

    v8f c00 = {}, c01 = {}, c10 = {}, c11 = {};
    for (int k = 0; k < D; k += 4) {
        int off = k + h2;
        v2f a0 = *(const v2f*)(arow0 + off);
        v2f a1 = *(const v2f*)(arow1 + off);
        v2f b0 = *(const v2f*)(brow0 + off);
        v2f b1 = *(const v2f*)(brow1 + off);
        c00 = __builtin_amdgcn_wmma_f32_16x16x4_f32(false, a0, false, b0, (short)0, c00, false, false);
        c01 = __builtin_amdgcn_wmma_f32_16x16x4_f32(false, a0, false, b1, (short)0, c01, false, false);
        c10 = __builtin_amdgcn_wmma_f32_16x16x4_f32(false, a1, false, b0, (short)0, c10, false, false);
        c11 = __builtin_amdgcn_wmma_f32_16x16x4_f32(false, a1, false, b1, (short)0, c11, false, false);
    }

    // Epilogue per C/D layout: VGPR r -> row r + 8*(lane>=16), col = lane&15.
    int ncol0 = n0 + l15;
    int ncol1 = n0 + 16 + l15;
    int mbase = m0 + ((lane >> 4) ? 8 : 0);
    float yb0 = y2[ncol0], yb1 = y2[ncol1];
    #pragma unroll
    for (int r = 0; r < 8; ++r) {
        int mr0 = mbase + r;
        int mr1 = mbase + 16 + r;
        float xa = x2[mr0], xb = x2[mr1];
        M[(size_t)mr0 * N + ncol0] = fmaxf(xa + yb0 - 2.0f * c00[r], 0.0f);
        M[(size_t)mr0 * N + ncol1] = fmaxf(xa + yb1 - 2.0f * c01[r], 0.0f);
        M[(size_t)mr1 * N + ncol0] = fmaxf(xb + yb0 - 2.0f * c10[r], 0.0f);
        M[(size_t)mr1 * N + ncol1] = fmaxf(xb + yb1 - 2.0f * c11[r], 0.0f);
    }
}

// ---------------- online logsumexp helpers ----------------------------------
__device__ __forceinline__ void lse_combine(float& m, float& s, float m2, float s2) {
    if (m2 > m) { s = s * __expf(m - m2) + s2; m = m2; }
    else        { s = s + s2 * __expf(m2 - m); }
}

// ---------------- f update, stage 1: partial LSE over 1024-col chunks -------
__global__ void row_lse_partial(const float* __restrict__ M, const float* __restrict__ g,
                                float2* __restrict__ part)
{
    __shared__ float sm[256], ss[256];
    int row = blockIdx.x;
    int c0  = blockIdx.y * 1024;
    int tid = threadIdx.x;
    const float* mr = M + (size_t)row * N + c0;
    float m = -INFINITY, s = 0.0f;
    #pragma unroll
    for (int k = 0; k < 4; ++k) {
        int c = tid + k * 256;
        lse_combine(m, s, g[c0 + c] - mr[c], 1.0f);
    }
    sm[tid] = m; ss[tid] = s;
    __syncthreads();
    for (int off = 128; off > 0; off >>= 1) {
        if (tid < off) {
            float mm = sm[tid], sc = ss[tid];
            lse_combine(mm, sc, sm[tid + off], ss[tid + off]);
            sm[tid] = mm; ss[tid] = sc;
        }
        __syncthreads();
    }
    if (tid == 0) part[(size_t)blockIdx.y * N + row] = make_float2(sm[0], ss[0]);
}

__global__ void row_lse_finish(const float2* __restrict__ part, float* __restrict__ f)
{
    int i = blockIdx.x * blockDim.x + threadIdx.x;   // N threads
    float m = -INFINITY, s = 0.0f;
    #pragma unroll
    for (int ch = 0; ch < 4; ++ch) {
        float2 p = part[(size_t)ch * N + i];
        lse_combine(m, s, p.x, p.y);
    }
    f[i] = -LOGN - (m + __logf(s));
}

// ---------------- g update, stage 1: 64 cols x 512-row chunk per block ------
__global__ void col_lse_partial(const float* __restrict__ M, const float* __restrict__ f,
                                float2* __restrict__ part)
{
    __shared__ float sm[256], ss[256];
    int tid = threadIdx.x;
    int tx  = tid & 63;            // column within group (coalesced loads)
    int rl  = tid >> 6;            // 0..3 row interleave
    int col = blockIdx.x * 64 + tx;
    int r0  = blockIdx.y * 512;
    float m = -INFINITY, s = 0.0f;
    for (int k = 0; k < 128; ++k) {
        int r = r0 + rl + 4 * k;
        lse_combine(m, s, f[r] - M[(size_t)r * N + col], 1.0f);
    }
    sm[tid] = m; ss[tid] = s;
    __syncthreads();
    if (tid < 64) {
        float mm = sm[tid], sc = ss[tid];
        lse_combine(mm, sc, sm[tid + 64],  ss[tid + 64]);
        lse_combine(mm, sc, sm[tid + 128], ss[tid + 128]);
        lse_combine(mm, sc, sm[tid + 192], ss[tid + 192]);
        part[(size_t)blockIdx.y * N + col] = make_float2(mm, sc);
    }
}

__global__ void col_lse_finish(const float2* __restrict__ part, float* __restrict__ g)
{
    int j = blockIdx.x * blockDim.x + threadIdx.x;   // N threads
    float m = -INFINITY, s = 0.0f;
    #pragma unroll
    for (int ch = 0; ch < 8; ++ch) {
        float2 p = part[(size_t)ch * N + j];
        lse_combine(m, s, p.x, p.y);
    }
    g[j] = -LOGN - (m + __logf(s));
}

__global__ void init_fg(float* __restrict__ f, float* __restrict__ g)
{
    int i = blockIdx.x * blockDim.x + threadIdx.x;
    f[i] = 0.0f;
    g[i] = 0.0f;
}

// ---------------- final objective ------------------------------------------
// value = sum_ij P*(M + logP + 2*logN - 1) + 1, with logP = f+g-M, P = exp(logP)
__global__ void final_row(const float* __restrict__ M, const float* __restrict__ f,
                          const float* __restrict__ g, float* __restrict__ rowsum)
{
    __shared__ float red[256];
    int row = blockIdx.x;
    int tid = threadIdx.x;
    float fi = f[row];
    const float* mr = M + (size_t)row * N;
    float acc = 0.0f;
    #pragma unroll 4
    for (int k = 0; k < 16; ++k) {
        int c = tid + k * 256;
        float Mv = mr[c];
        float lp = fi + g[c] - Mv;
        float p  = __expf(lp);
        acc += p * (Mv + lp + (2.0f * LOGN - 1.0f));
    }
    red[tid] = acc;
    __syncthreads();
    for (int off = 128; off > 0; off >>= 1) {
        if (tid < off) red[tid] += red[tid + off];
        __syncthreads();
    }
    if (tid == 0) rowsum[row] = red[0];
}

__global__ void final_reduce(const float* __restrict__ rowsum, float* __restrict__ out)
{
    __shared__ float red[256];
    int tid = threadIdx.x;
    float acc = 0.0f;
    #pragma unroll
    for (int k = 0; k < 16; ++k) acc += rowsum[tid + k * 256];
    red[tid] = acc;
    __syncthreads();
    for (int off = 128; off > 0; off >>= 1) {
        if (tid < off) red[tid] += red[tid + off];
        __syncthreads();
    }
    if (tid == 0) out[0] = red[0] + 1.0f;
}

extern "C" void kernel_launch(void* const* d_in, const int* in_sizes, int n_in,
                              void* d_out, int out_size, void* d_ws, size_t ws_size,
                              hipStream_t stream)
{
    (void)in_sizes; (void)n_in; (void)out_size; (void)ws_size;
    const float* X = (const float*)d_in[0];   // source [4096,512]
    const float* Y = (const float*)d_in[1];   // target [4096,512]
    float* out = (float*)d_out;

    // workspace layout (all 16KB-aligned slices)
    float*  M      = (float*)d_ws;                    // N*N floats (64 MB, L2-resident)
    float*  f      = M + (size_t)N * N;               // N
    float*  g      = f + N;                           // N
    float*  x2     = g + N;                           // N
    float*  y2     = x2 + N;                          // N
    float2* rowp   = (float2*)(y2 + N);               // 4*N float2
    float2* colp   = rowp + 4 * (size_t)N;            // 8*N float2
    float*  rowsum = (float*)(colp + 8 * (size_t)N);  // N

    sqnorm_kernel<<<N, 256, 0, stream>>>(X, x2);
    sqnorm_kernel<<<N, 256, 0, stream>>>(Y, y2);
    cost_gemm_kernel<<<dim3(N / 128, N / 64), 256, 0, stream>>>(X, Y, x2, y2, M);
    init_fg<<<16, 256, 0, stream>>>(f, g);

    for (int it = 0; it < 100; ++it) {
        row_lse_partial<<<dim3(N, 4), 256, 0, stream>>>(M, g, rowp);
        row_lse_finish<<<16, 256, 0, stream>>>(rowp, f);
        col_lse_partial<<<dim3(N / 64, 8), 256, 0, stream>>>(M, f, colp);
        col_lse_finish<<<16, 256, 0, stream>>>(colp, g);
    }

    final_row<<<N, 256, 0, stream>>>(M, f, g, rowsum);
    final_reduce<<<1, 256, 0, stream>>>(rowsum, out);
}